// LengthRegulator_1932735283443
// MI455X (gfx1250) — compile-verified
//
#include <hip/hip_runtime.h>
#include <hip/hip_bf16.h>

// ---------------------------------------------------------------------------
// LengthRegulator for MI455X (gfx1250).
// Pure bandwidth problem: ~252 MB of HBM traffic -> ~10.8 us at 23.3 TB/s.
// No FLOPs -> no WMMA. CDNA5-specific path used: async global<->LDS copies
// (GLOBAL_LOAD_ASYNC_TO_LDS_B128 / GLOBAL_STORE_ASYNC_FROM_LDS_B128, tracked
// by ASYNCcnt), with a direct b128 fallback if builtins are absent.
// ---------------------------------------------------------------------------

constexpr int B_ = 64;
constexpr int S_ = 512;
constexpr int D_ = 384;
constexpr int T_ = 2048;          // max_len (scalar input, fixed by harness)
constexpr int NV = D_ / 4;        // 96 float4 per row
constexpr int ROWS = 8;           // rows (waves) per block in gather kernel
constexpr int THREADS2 = ROWS * 32;
constexpr int CHUNKS = NV / 32;   // 3 float4 per lane per row

#if __has_builtin(__builtin_amdgcn_global_load_async_to_lds_b128) && \
    __has_builtin(__builtin_amdgcn_global_store_async_from_lds_b128) && \
    __has_builtin(__builtin_amdgcn_s_wait_asynccnt)
#define LR_ASYNC_LDS 1
#else
#define LR_ASYNC_LDS 0
#endif

#if LR_ASYNC_LDS
// Builtin parameter types per hipcc diagnostic:
//   arg0: int __attribute__((vector_size(16))) __device__(*)   (global, non-const)
//   arg1: LDS pointer (shared address space)
typedef int lr_v4i __attribute__((vector_size(16)));
typedef __attribute__((address_space(1))) lr_v4i* lr_gp;   // global v4i*
typedef __attribute__((address_space(3))) lr_v4i* lr_lp;   // LDS v4i*
#endif

// ---------------------------------------------------------------------------
// Kernel 1: per-batch duration scan + frame->phoneme map + lengths.
// One block per batch (512 threads = 16 waves). LDS inclusive scan.
// ---------------------------------------------------------------------------
__global__ __launch_bounds__(S_) void lr_scan_map(const int* __restrict__ dur,
                                                  int* __restrict__ map,
                                                  float* __restrict__ lengths_out) {
    const int b = blockIdx.x;
    const int s = threadIdx.x;

    __shared__ int sm[S_];

    int d = dur[b * S_ + s];
    d = d < 0 ? 0 : (d > 300 ? 300 : d);   // clamp(0, 300)
    sm[s] = d;
    __syncthreads();

    // Hillis-Steele inclusive scan over S_=512 (9 steps, ds_ ops).
    #pragma unroll
    for (int off = 1; off < S_; off <<= 1) {
        int v = 0;
        if (s >= off) v = sm[s - off];
        __syncthreads();
        if (s >= off) sm[s] += v;
        __syncthreads();
    }

    // Initialize this batch's frame map to -1 (masked / zero frames).
    #pragma unroll
    for (int k = 0; k < T_ / S_; ++k)
        map[b * T_ + k * S_ + s] = -1;
    __syncthreads();   // order init before scatter (block-scope global fence)

    // Scatter: phoneme s covers frames [csum[s-1], min(csum[s], T_)).
    int end = sm[s];
    int start = (s == 0) ? 0 : sm[s - 1];
    if (end > T_) end = T_;
    for (int t = start; t < end; ++t)
        map[b * T_ + t] = s;

    if (s == S_ - 1) {
        int len = sm[S_ - 1];
        if (len > T_) len = T_;
        lengths_out[b] = (float)len;   // int value <= 2048, exact in f32
    }
}

// ---------------------------------------------------------------------------
// Kernel 2: bandwidth-bound row gather. One wave32 per output row:
// 96 float4 per row, 3 float4 per lane -> gather index is wave-uniform,
// so branches keep EXEC all-1s within each wave.
// ---------------------------------------------------------------------------
__global__ __launch_bounds__(THREADS2) void lr_gather(const float4* __restrict__ x,
                                                      const int* __restrict__ map,
                                                      float4* __restrict__ out) {
    const int tid  = threadIdx.x;
    const int wave = tid >> 5;
    const int lane = tid & 31;
    const int row  = blockIdx.x * ROWS + wave;   // row in [0, B*T)
    const int b    = row / T_;

    const int idx = map[row];                    // wave-uniform gather index
    float4* dst = out + (size_t)row * NV;

    if (idx < 0) {
        // Masked frame: write zeros (full overwrite of poisoned d_out).
        const float4 z = make_float4(0.f, 0.f, 0.f, 0.f);
        #pragma unroll
        for (int k = 0; k < CHUNKS; ++k)
            dst[lane + 32 * k] = z;
        return;
    }

    const float4* src = x + ((size_t)b * S_ + idx) * NV;

#if LR_ASYNC_LDS
    // CDNA5 async data-mover path: global -> LDS -> global, tracked by
    // ASYNCcnt. Each wave only consumes LDS it produced, so a per-wave
    // s_wait_asynccnt(0) between the load and store phases is sufficient.
    __shared__ float4 buf[THREADS2 * CHUNKS];
    #pragma unroll
    for (int k = 0; k < CHUNKS; ++k)
        __builtin_amdgcn_global_load_async_to_lds_b128(
            (lr_gp)(src + lane + 32 * k),
            (lr_lp)&buf[k * THREADS2 + tid],
            0, 0);
    __builtin_amdgcn_s_wait_asynccnt(0);
    #pragma unroll
    for (int k = 0; k < CHUNKS; ++k)
        __builtin_amdgcn_global_store_async_from_lds_b128(
            (lr_gp)(dst + lane + 32 * k),
            (lr_lp)&buf[k * THREADS2 + tid],
            0, 0);
    // Outstanding async stores drain at s_endpgm (implicit wait-idle).
#else
    // Fallback: direct 128-bit copy (also roofline-optimal).
    #pragma unroll
    for (int k = 0; k < CHUNKS; ++k)
        dst[lane + 32 * k] = src[lane + 32 * k];
#endif
}

// ---------------------------------------------------------------------------
// Launch
// ---------------------------------------------------------------------------
extern "C" void kernel_launch(void* const* d_in, const int* in_sizes, int n_in,
                              void* d_out, int out_size, void* d_ws, size_t ws_size,
                              hipStream_t stream) {
    const float* x   = (const float*)d_in[0];   // [B, S, D] f32
    const int*   dur = (const int*)d_in[1];     // [B, S] i32
    // d_in[2] = max_len scalar (2048), baked into T_.

    float* out     = (float*)d_out;                       // [B, T, D] f32 ...
    float* lengths = out + (size_t)B_ * T_ * D_;          // ... then [B] lengths
    int*   map     = (int*)d_ws;                          // [B, T] scratch (512 KB)

    lr_scan_map<<<B_, S_, 0, stream>>>(dur, map, lengths);
    lr_gather<<<(B_ * T_) / ROWS, THREADS2, 0, stream>>>(
        (const float4*)x, map, (float4*)out);
}